// Model_88167088652800
// MI455X (gfx1250) — compile-verified
//
#include <hip/hip_runtime.h>

// Problem constants (match reference).
#define NU   100000
#define NI   50000
#define NE   1000000
#define DD   64
#define OUTD 256            // 4 levels * 64, output concat stride
#define LDS_COL_STRIDE 132  // padded column stride (floats) -> conflict-free b-frag reads

typedef float v2f __attribute__((ext_vector_type(2)));
typedef float v8f __attribute__((ext_vector_type(8)));

// ---------------------------------------------------------------------------
// Edge message pass, fused:  msg = norm_e * ( [x_s | x_s*x_d] @ [W1;W2] + (b1+b2) )
// then scatter-add into hacc[dst].  One wave = 16 edges; 128 f32 WMMAs / wave.
// ---------------------------------------------------------------------------
__global__ __launch_bounds__(128) void edge_pass_kernel(
    const float* __restrict__ xsrc,   // [Nsrc, OUTD] current-level block (col offset pre-applied)
    const float* __restrict__ xdst,   // [Ndst, OUTD]
    const float* __restrict__ w1,     // [64,64] row-major
    const float* __restrict__ w2,     // [64,64] row-major
    const float* __restrict__ b1,     // [64]
    const float* __restrict__ b2,     // [64]
    const float* __restrict__ norm,   // [E]
    const int*   __restrict__ esrc,   // [E]
    const int*   __restrict__ edst,   // [E]
    float*       __restrict__ hacc,   // [Ndst, 64] accumulator
    int nE)
{
    __shared__ float ldsW[64 * LDS_COL_STRIDE]; // combined [W1;W2], column-major, padded
    __shared__ float sBias[64];
    __shared__ int   sSrc[4][16];
    __shared__ int   sDst[4][16];
    __shared__ float sNrm[4][16];

    const int tid = threadIdx.x;

    // Stage combined weight matrix [W1;W2] (128 x 64) column-major into LDS.
    for (int i = tid; i < 4096; i += 128) {
        int r = i >> 6, c = i & 63;
        ldsW[c * LDS_COL_STRIDE + r]      = w1[i];
        ldsW[c * LDS_COL_STRIDE + 64 + r] = w2[i];
    }
    if (tid < 64) sBias[tid] = b1[tid] + b2[tid];

    const int wv   = tid >> 5;      // wave in block (4)
    const int lane = tid & 31;
    const int lm   = lane & 15;     // A-row (edge) index, and B/C column lane
    const int hi   = lane >> 4;     // upper half-wave flag
    const int koff = hi * 2;        // K sub-offset per f32 WMMA A/B layout

    // Edge metadata: 16 edges per wave.  Clamp tail, zero its norm -> zero message,
    // keeps EXEC all-ones for WMMA.
    if (lane < 16) {
        int e  = blockIdx.x * 64 + wv * 16 + lane;
        int ec = (e < nE) ? e : (nE - 1);
        sSrc[wv][lane] = esrc[ec];
        sDst[wv][lane] = edst[ec];
        sNrm[wv][lane] = (e < nE) ? norm[ec] : 0.0f;
    }
    __syncthreads();

    const float* xs = xsrc + (long)sSrc[wv][lm] * OUTD;
    const float* xd = xdst + (long)sDst[wv][lm] * OUTD;

    // Per-lane epilogue metadata (8 edge rows per lane), cached in registers
    // and reused across all 4 n-tiles.
    int   dstm[8];
    float nrmm[8];
#pragma unroll
    for (int v = 0; v < 8; ++v) {
        dstm[v] = sDst[wv][hi * 8 + v];
        nrmm[v] = sNrm[wv][hi * 8 + v];
    }

    // Preload all 32 A fragments (K = 128 combined).
    v2f afrag[32];
#pragma unroll
    for (int kk = 0; kk < 16; ++kk) {          // first 64 features: x_src (W1 term)
        int f = 4 * kk + koff;
        afrag[kk] = *(const v2f*)(xs + f);
    }
#pragma unroll
    for (int kk = 16; kk < 32; ++kk) {         // next 64 features: x_src * x_dst (W2 term)
        int f = 4 * kk + koff - 64;
        v2f u = *(const v2f*)(xs + f);
        v2f w = *(const v2f*)(xd + f);
        afrag[kk] = u * w;
    }

    // 4 N-tiles of 16 columns; 32 K-steps each.
    const float* wcol = &ldsW[lm * LDS_COL_STRIDE + koff];
    for (int n = 0; n < 4; ++n) {
        // Bulk-preload all 32 B fragments for this n-tile so the ds_loads
        // overlap the WMMA chain instead of gating each WMMA pair.
        v2f bfrag[32];
#pragma unroll
        for (int kk = 0; kk < 32; ++kk)
            bfrag[kk] = *(const v2f*)(wcol + 4 * kk);

        v8f acc = {};
#pragma unroll
        for (int kk = 0; kk < 32; ++kk)
            acc = __builtin_amdgcn_wmma_f32_16x16x4_f32(
                false, afrag[kk], false, bfrag[kk], (short)0, acc, false, false);

        const int   col = 16 * n + lm;
        const float bb  = sBias[col];
#pragma unroll
        for (int v = 0; v < 8; ++v) {
            float msg = nrmm[v] * (acc[v] + bb);
            __hip_atomic_fetch_add(hacc + (long)dstm[v] * DD + col, msg,
                                   __ATOMIC_RELAXED, __HIP_MEMORY_SCOPE_AGENT);
        }
        wcol += 16 * LDS_COL_STRIDE;   // advance to next 16-column tile
    }
}

// ---------------------------------------------------------------------------
// h -> leaky_relu(0.2) -> row L2-normalize -> write into out column block.
// One wave per row (lane holds elements lane and lane+32).
// ---------------------------------------------------------------------------
__global__ __launch_bounds__(256) void finalize_kernel(
    const float* __restrict__ h, float* __restrict__ out, int nrows, int colbase)
{
    const int wv   = threadIdx.x >> 5;
    const int lane = threadIdx.x & 31;
    const long row = (long)blockIdx.x * 8 + wv;
    if (row >= nrows) return;

    float v0 = h[row * DD + lane];
    float v1 = h[row * DD + 32 + lane];
    v0 = (v0 > 0.0f) ? v0 : 0.2f * v0;
    v1 = (v1 > 0.0f) ? v1 : 0.2f * v1;

    float s = v0 * v0 + v1 * v1;
#pragma unroll
    for (int o = 16; o > 0; o >>= 1) s += __shfl_xor(s, o, 32);

    float inv = 1.0f / fmaxf(sqrtf(s), 1e-12f);
    out[row * OUTD + colbase + lane]      = v0 * inv;
    out[row * OUTD + colbase + 32 + lane] = v1 * inv;
}

// Copy raw embeddings into level-0 column block of the concat output.
__global__ void init_copy_kernel(const float* __restrict__ emb,
                                 float* __restrict__ out, long n)
{
    long i = (long)blockIdx.x * 256 + threadIdx.x;
    if (i < n) {
        long r = i >> 6;
        int  c = (int)(i & 63);
        out[r * OUTD + c] = emb[i];
    }
}

__global__ void zero_kernel(float* __restrict__ p, long n)
{
    long i = (long)blockIdx.x * 256 + threadIdx.x;
    if (i < n) p[i] = 0.0f;
}

// ---------------------------------------------------------------------------
extern "C" void kernel_launch(void* const* d_in, const int* in_sizes, int n_in,
                              void* d_out, int out_size, void* d_ws, size_t ws_size,
                              hipStream_t stream)
{
    const float* user_emb = (const float*)d_in[0];
    const float* item_emb = (const float*)d_in[1];
    const float* W1       = (const float*)d_in[2];
    const float* b1       = (const float*)d_in[3];
    const float* W2       = (const float*)d_in[4];
    const float* b2       = (const float*)d_in[5];
    const float* norm_ui  = (const float*)d_in[6];
    const float* norm_iu  = (const float*)d_in[7];
    const int*   ui_src   = (const int*)d_in[8];
    const int*   ui_dst   = (const int*)d_in[9];
    const int*   iu_src   = (const int*)d_in[10];
    const int*   iu_dst   = (const int*)d_in[11];

    float* out_u = (float*)d_out;                       // [NU, 256]
    float* out_i = out_u + (long)NU * OUTD;             // [NI, 256]
    float* h_u   = (float*)d_ws;                        // [NU, 64]
    float* h_i   = h_u + (long)NU * DD;                 // [NI, 64]

    // Level 0 = raw embeddings into cols [0,64).
    init_copy_kernel<<<(NU * DD + 255) / 256, 256, 0, stream>>>(user_emb, out_u, (long)NU * DD);
    init_copy_kernel<<<(NI * DD + 255) / 256, 256, 0, stream>>>(item_emb, out_i, (long)NI * DD);

    const int edgeBlocks = (NE + 63) / 64;
    const long hTotal    = (long)(NU + NI) * DD;

    for (int l = 0; l < 3; ++l) {
        zero_kernel<<<(int)((hTotal + 255) / 256), 256, 0, stream>>>(h_u, hTotal);

        const float* xu = out_u + l * DD;   // current level block, row stride 256
        const float* xi = out_i + l * DD;
        const float* w1l = W1 + (long)l * DD * DD;
        const float* w2l = W2 + (long)l * DD * DD;
        const float* b1l = b1 + (long)l * DD;
        const float* b2l = b2 + (long)l * DD;

        // user -> item messages accumulate into h_i
        edge_pass_kernel<<<edgeBlocks, 128, 0, stream>>>(
            xu, xi, w1l, w2l, b1l, b2l, norm_ui, ui_src, ui_dst, h_i, NE);
        // item -> user messages accumulate into h_u
        edge_pass_kernel<<<edgeBlocks, 128, 0, stream>>>(
            xi, xu, w1l, w2l, b1l, b2l, norm_iu, iu_src, iu_dst, h_u, NE);

        finalize_kernel<<<(NU + 7) / 8, 256, 0, stream>>>(h_u, out_u, NU, (l + 1) * DD);
        finalize_kernel<<<(NI + 7) / 8, 256, 0, stream>>>(h_i, out_i, NI, (l + 1) * DD);
    }
}